// KFactor_549755814467
// MI455X (gfx1250) — compile-verified
//
#include <hip/hip_runtime.h>
#include <hip/hip_bf16.h>

// Problem constants (from reference): B=16384 samples, N=64 clusters,
// F=128 feature dim, K=64 dictionary atoms.
#define BB 16384
#define NN 64
#define FF 128
#define KK 64
#define BT 128   // samples per workgroup tile in kproj

typedef __attribute__((ext_vector_type(2))) float v2f;
typedef __attribute__((ext_vector_type(4))) float f4;
typedef __attribute__((ext_vector_type(8))) float v8f;

typedef __attribute__((ext_vector_type(4))) unsigned int u32x4;
typedef __attribute__((ext_vector_type(4))) int i32x4;
typedef __attribute__((ext_vector_type(8))) int i32x8;

// ---------------------------------------------------------------------------
// Tensor Data Mover: 1-row 2D tile of `n8` 8-byte elements, contiguous in
// global memory -> contiguous in LDS. Descriptor per CDNA5 ISA §8.3/8.4:
//   group0: count=1 | lds_addr | global_addr[56:0] | type=2
//   group1: data_size=3 (8B), tensor_dim0=tile_dim0=n8, tensor_dim1=tile_dim1=1,
//           tensor_dim0_stride=n8, workgroup_mask=0 (not in a cluster)
//   groups 2/3 (and group4 on clang-23): zero (<=2D tensor)
// Issued once per wave (EXEC-independent); caller guards to a single wave and
// waits with s_wait_tensorcnt 0 before a workgroup barrier.
// ---------------------------------------------------------------------------
__device__ static inline void tdm_load_1d(unsigned lds_addr, const void* gptr,
                                          unsigned n8) {
    const unsigned long long ga = (unsigned long long)gptr;
    u32x4 g0;
    g0.x = 1u;                                          // count=1, user D#
    g0.y = lds_addr;                                    // LDS byte address
    g0.z = (unsigned)ga;                                // global_addr[31:0]
    g0.w = (unsigned)((ga >> 32) & 0x01FFFFFFull)       // global_addr[56:32]
         | 0x80000000u;                                 // type=2 ("image")
    i32x8 g1;
    g1[0] = (int)(3u << 16);                            // data_size = 8 bytes
    g1[1] = (int)((n8 & 0xFFFFu) << 16);                // tensor_dim0[15:0]
    g1[2] = (int)(((n8 >> 16) & 0xFFFFu) | (1u << 16)); // dim0[31:16] | tensor_dim1=1
    g1[3] = (int)((n8 & 0xFFFFu) << 16);                // tile_dim0 = n8
    g1[4] = 1;                                          // tile_dim1 = 1
    g1[5] = (int)n8;                                    // tensor_dim0_stride[31:0]
    g1[6] = 0;                                          // stride[47:32] | dim1_stride
    g1[7] = 0;
    i32x4 gz = {0, 0, 0, 0};
#if __clang_major__ >= 23
    i32x8 gz8 = {0, 0, 0, 0, 0, 0, 0, 0};
    __builtin_amdgcn_tensor_load_to_lds(g0, g1, gz, gz, gz8, 0);
#else
    __builtin_amdgcn_tensor_load_to_lds(g0, g1, gz, gz, 0);
#endif
}

__device__ static inline unsigned lds_off(const void* p) {
    // Flat shared-aperture address: low 32 bits are the LDS byte offset.
    return (unsigned)(unsigned long long)p;
}

// ---------------------------------------------------------------------------
// Kernel 1: DTD[n] = D_n^T D_n   (64 x 64 per cluster; tiny, VALU is fine)
// Also zero-initializes the global loss accumulator (d_out[0]).
// ---------------------------------------------------------------------------
__global__ __launch_bounds__(256) void kdtd(const float* __restrict__ D,
                                            float* __restrict__ DTD,
                                            float* __restrict__ loss) {
    __shared__ float Ds[FF * KK];          // 32 KB
    const int n = blockIdx.x, tid = threadIdx.x;
    if (n == 0 && tid == 0) *loss = 0.0f;
    for (int e = tid; e < FF * KK; e += 256) Ds[e] = D[(size_t)n * FF * KK + e];
    __syncthreads();
    for (int e = tid; e < KK * KK; e += 256) {
        const int j = e >> 6, k = e & 63;
        float s = 0.0f;
        #pragma unroll 8
        for (int i = 0; i < FF; ++i) s = fmaf(Ds[i * KK + j], Ds[i * KK + k], s);
        DTD[(size_t)n * KK * KK + e] = s;
    }
}

// ---------------------------------------------------------------------------
// Kernel 2: INV[n] = inv(DTD[n]) via Gauss-Jordan in LDS, one block per n.
// DTD is a Wishart matrix of a dense Gaussian dictionary -> well conditioned.
// ---------------------------------------------------------------------------
__global__ __launch_bounds__(256) void kinv(const float* __restrict__ DTD,
                                            float* __restrict__ INV) {
    __shared__ float A[KK * KK];
    __shared__ float M[KK * KK];
    __shared__ float fac[KK];
    __shared__ float piv;
    const int n = blockIdx.x, tid = threadIdx.x;
    for (int e = tid; e < KK * KK; e += 256) {
        A[e] = DTD[(size_t)n * KK * KK + e];
        M[e] = ((e >> 6) == (e & 63)) ? 1.0f : 0.0f;
    }
    __syncthreads();
    for (int c = 0; c < KK; ++c) {
        if (tid == 0) piv = 1.0f / A[c * KK + c];
        __syncthreads();
        if (tid < KK)            A[c * KK + tid]        *= piv;
        else if (tid < 2 * KK)   M[c * KK + (tid - KK)] *= piv;
        __syncthreads();
        if (tid < KK) fac[tid] = (tid == c) ? 0.0f : A[tid * KK + c];
        __syncthreads();
        for (int e = tid; e < KK * KK; e += 256) {
            const int r = e >> 6, cc = e & 63;
            if (r != c) {
                const float f = fac[r];
                A[r * KK + cc] = fmaf(-f, A[c * KK + cc], A[r * KK + cc]);
                M[r * KK + cc] = fmaf(-f, M[c * KK + cc], M[r * KK + cc]);
            }
        }
        __syncthreads();
    }
    for (int e = tid; e < KK * KK; e += 256) INV[(size_t)n * KK * KK + e] = M[e];
}

// ---------------------------------------------------------------------------
// Kernel 3: the WMMA projection pipeline. One block = (cluster n, 128-sample
// tile). LDS: X(64K) D(32K) D^T(32K) INV(16K) DTX(32K) CS(32K) = 208 KB.
// Staging via Tensor Data Mover; three chained f32 GEMMs on the matrix pipe.
//
// f32 WMMA fragment layouts (ISA 7.12.2):
//   A 16x4 : lanes 0-15 row M=lane  {v0=K0,v1=K1}; lanes 16-31 {v0=K2,v1=K3}
//   B 4x16 : lanes 0-15 col N=lane  {v0=row K0,v1=row K1}; half 1 {K2,K3}
//   C 16x16: VGPR g = rows {g, g+8} split across lane halves, N = lane&15
// ---------------------------------------------------------------------------
__global__ __launch_bounds__(256) void kproj(const float* __restrict__ x,
                                             const float* __restrict__ D,
                                             const float* __restrict__ INV,
                                             float* __restrict__ norms) {
    extern __shared__ float sm[];
    float* Xs  = sm;                 // BT x FF  (row stride 128)
    float* Ds  = Xs  + BT * FF;      // FF x KK  (row stride 64)   D_n[i][j]
    float* DsT = Ds  + FF * KK;      // KK x FF  (row stride 128)  D_n^T[j][i]
    float* IV  = DsT + KK * FF;      // KK x KK
    float* DTX = IV  + KK * KK;      // BT x KK
    float* CS  = DTX + BT * KK;      // BT x KK

    const int n  = blockIdx.x;
    const int b0 = blockIdx.y * BT;
    const int tid = threadIdx.x;

    // --- TDM staging: three contiguous tiles, issued by wave 0 only --------
    if (tid < 32) {
        tdm_load_1d(lds_off(Xs), x + (size_t)b0 * FF, (BT * FF) / 2);
        tdm_load_1d(lds_off(Ds), D + (size_t)n * FF * KK, (FF * KK) / 2);
        tdm_load_1d(lds_off(IV), INV + (size_t)n * KK * KK, (KK * KK) / 2);
        __builtin_amdgcn_s_wait_tensorcnt(0);
    }
    __syncthreads();

    // One-time transpose so GEMM1 B-fragments are contiguous ds_load_b64s.
    for (int e = tid; e < FF * KK; e += 256) {
        const int i = e >> 6, j = e & 63;
        DsT[j * FF + i] = Ds[e];
    }
    __syncthreads();

    const int lane = tid & 31;
    const int m16  = lane & 15;     // row (A/C) or col (B) within 16-lane half
    const int half = lane >> 4;     // which half of the wave
    const int mrow = (tid >> 5) * 16;  // each wave owns one 16-row M strip

    // ---- GEMM 1: DTX[BTxKK] = Xs[BTxFF] @ Ds[FFxKK]  (32 K-steps of 4) ----
    {
        v8f acc[4] = {};
        for (int kk = 0; kk < FF / 4; ++kk) {
            const int k0 = kk * 4;
            v2f a;
            a.x = Xs[(mrow + m16) * FF + k0 + 2 * half];
            a.y = Xs[(mrow + m16) * FF + k0 + 2 * half + 1];
            #pragma unroll
            for (int ni = 0; ni < 4; ++ni) {
                v2f b;   // B[k][n] = D_n[k][n] = DsT[n][k] (contiguous in k)
                b.x = DsT[(ni * 16 + m16) * FF + k0 + 2 * half];
                b.y = DsT[(ni * 16 + m16) * FF + k0 + 2 * half + 1];
                acc[ni] = __builtin_amdgcn_wmma_f32_16x16x4_f32(
                    false, a, false, b, (short)0, acc[ni], false, false);
            }
        }
        #pragma unroll
        for (int ni = 0; ni < 4; ++ni)
            #pragma unroll
            for (int g = 0; g < 8; ++g)
                DTX[(mrow + g + 8 * half) * KK + ni * 16 + m16] = acc[ni][g];
    }
    __syncthreads();

    // ---- GEMM 2: CS = DTX @ INV^T   (Cs[b,k] = sum_j Dtx[b,j] * inv[k,j]) --
    {
        v8f acc[4] = {};
        for (int kk = 0; kk < KK / 4; ++kk) {
            const int k0 = kk * 4;
            v2f a;
            a.x = DTX[(mrow + m16) * KK + k0 + 2 * half];
            a.y = DTX[(mrow + m16) * KK + k0 + 2 * half + 1];
            #pragma unroll
            for (int ni = 0; ni < 4; ++ni) {
                v2f b;   // B[j][k] = inv[k][j] -> row of IV, contiguous in j
                b.x = IV[(ni * 16 + m16) * KK + k0 + 2 * half];
                b.y = IV[(ni * 16 + m16) * KK + k0 + 2 * half + 1];
                acc[ni] = __builtin_amdgcn_wmma_f32_16x16x4_f32(
                    false, a, false, b, (short)0, acc[ni], false, false);
            }
        }
        #pragma unroll
        for (int ni = 0; ni < 4; ++ni)
            #pragma unroll
            for (int g = 0; g < 8; ++g)
                CS[(mrow + g + 8 * half) * KK + ni * 16 + m16] = acc[ni][g];
    }
    __syncthreads();

    // ---- GEMM 3: x_hat = CS @ Ds^T ; fuse residual ||x_hat - x||^2 --------
    // kk-outer with 8 live accumulators: each CS A-fragment feeds 8 WMMAs.
    {
        v8f acc[8] = {};
        for (int kk = 0; kk < KK / 4; ++kk) {
            const int k0 = kk * 4;
            v2f a;
            a.x = CS[(mrow + m16) * KK + k0 + 2 * half];
            a.y = CS[(mrow + m16) * KK + k0 + 2 * half + 1];
            #pragma unroll
            for (int ni = 0; ni < 8; ++ni) {
                v2f b;   // B[j][i] = D[i][j] -> row of Ds, contiguous in j
                b.x = Ds[(ni * 16 + m16) * KK + k0 + 2 * half];
                b.y = Ds[(ni * 16 + m16) * KK + k0 + 2 * half + 1];
                acc[ni] = __builtin_amdgcn_wmma_f32_16x16x4_f32(
                    false, a, false, b, (short)0, acc[ni], false, false);
            }
        }
        // Residual: row (g,half) of this wave's strip lives in one 16-lane
        // half; columns span lanes (m16) x tiles (ni). Butterfly-reduce the
        // 16 lanes, then lane m16==0 stores the row's squared norm directly.
        float rloc[8];
        #pragma unroll
        for (int g = 0; g < 8; ++g) rloc[g] = 0.0f;
        #pragma unroll
        for (int ni = 0; ni < 8; ++ni)
            #pragma unroll
            for (int g = 0; g < 8; ++g) {
                const float xv = Xs[(mrow + g + 8 * half) * FF + ni * 16 + m16];
                const float d  = acc[ni][g] - xv;
                rloc[g] = fmaf(d, d, rloc[g]);
            }
        #pragma unroll
        for (int g = 0; g < 8; ++g) {
            float s = rloc[g];
            s += __shfl_xor(s, 1);
            s += __shfl_xor(s, 2);
            s += __shfl_xor(s, 4);
            s += __shfl_xor(s, 8);
            if (m16 == 0)
                norms[(size_t)n * BB + b0 + mrow + g + 8 * half] = s;
        }
    }
}

// ---------------------------------------------------------------------------
// Kernel 4: per-sample argmin over clusters + loss = sum of min sq-residual.
// (x_rec == x_hat[label], so loss is exactly the sum of per-sample minima.)
// ---------------------------------------------------------------------------
__global__ __launch_bounds__(256) void kargmin(const float* __restrict__ norms,
                                               float* __restrict__ loss,
                                               int* __restrict__ label) {
    const int b = blockIdx.x * 256 + threadIdx.x;
    float best = norms[b];
    int bi = 0;
    for (int nn = 1; nn < NN; ++nn) {
        const float v = norms[(size_t)nn * BB + b];
        if (v < best) { best = v; bi = nn; }   // strict < keeps first index (jnp tie rule)
    }
    label[b] = bi;
    __shared__ float red[256];
    red[threadIdx.x] = best;
    __syncthreads();
    for (int s = 128; s > 0; s >>= 1) {
        if (threadIdx.x < s) red[threadIdx.x] += red[threadIdx.x + s];
        __syncthreads();
    }
    if (threadIdx.x == 0) atomicAdd(loss, red[0]);
}

// ---------------------------------------------------------------------------
extern "C" void kernel_launch(void* const* d_in, const int* in_sizes, int n_in,
                              void* d_out, int out_size, void* d_ws, size_t ws_size,
                              hipStream_t stream) {
    const float* x = (const float*)d_in[0];   // [B,F]
    const float* D = (const float*)d_in[1];   // [N,F,K]

    // Workspace layout: DTD (1MB) | INV (1MB) | norms (4MB)
    float* DTD   = (float*)d_ws;
    float* INV   = DTD + (size_t)NN * KK * KK;
    float* norms = INV + (size_t)NN * KK * KK;

    float* loss  = (float*)d_out;             // d_out[0]  = se_loss
    int*   label = ((int*)d_out) + 1;         // d_out[1:] = labels

    kdtd<<<NN, 256, 0, stream>>>(D, DTD, loss);
    kinv<<<NN, 256, 0, stream>>>(DTD, INV);

    const size_t smem = (size_t)(BT * FF + FF * KK + KK * FF + KK * KK +
                                 BT * KK + BT * KK) * sizeof(float); // 208 KB
    hipFuncSetAttribute(reinterpret_cast<const void*>(kproj),
                        hipFuncAttributeMaxDynamicSharedMemorySize, (int)smem);
    kproj<<<dim3(NN, BB / BT), 256, smem, stream>>>(x, D, INV, norms);

    kargmin<<<BB / 256, 256, 0, stream>>>(norms, loss, label);
}